// ROIViTExtractor_26568667693340
// MI455X (gfx1250) — compile-verified
//
#include <hip/hip_runtime.h>

typedef __attribute__((ext_vector_type(2))) float v2f;
typedef __attribute__((ext_vector_type(8))) float v8f;

#define NSIDE   14
#define NPATCH  196     // 14*14, K dim
#define HID     768     // N dim
#define NROI    32      // M dim
#define NTOK    33      // NROI + 1
#define BLK     256     // 8 waves (wave32)
#define NSPLIT  2       // blocks per frame (N-dim split for occupancy)

// One block: frame b = blockIdx.x / NSPLIT, N-half s = blockIdx.x % NSPLIT.
// Computes out[b, 1+m, d] = (1/196) * sum_p ratio[m,p] * pt[b,p,d] for 24 N-tiles,
// via V_WMMA_F32_16X16X4_F32. Split 0 also copies the cls row.
__global__ __launch_bounds__(BLK) void roi_vit_wmma_kernel(
    const float* __restrict__ cls,    // (B, 768)
    const float* __restrict__ pt,     // (B, 196, 768)
    const float* __restrict__ bbox,   // (B, 32, 4)
    float* __restrict__ out)          // (B, 33, 768)
{
    __shared__ float sR[NROI * NPATCH];   // 25088 B: ratio matrix, row-major (M x K)

    const int b   = blockIdx.x / NSPLIT;
    const int s   = blockIdx.x - b * NSPLIT;
    const int tid = threadIdx.x;

    // ---- cls row (split 0 only) ----
    if (s == 0) {
        const float* c = cls + (size_t)b * HID;
        float*       o = out + (size_t)b * NTOK * HID;
        for (int d = tid; d < HID; d += BLK) o[d] = c[d];
    }

    // ---- stage 1: ratio matrix into LDS ----
    {
        const float* bbb = bbox + (size_t)b * NROI * 4;
        for (int idx = tid; idx < NROI * NPATCH; idx += BLK) {
            const int n  = idx / NPATCH;
            const int p  = idx - n * NPATCH;
            const int pr = p / NSIDE;
            const int pc = p - pr * NSIDE;
            const float px1 = (float)(pc * 16);
            const float py1 = (float)(pr * 16);
            const float bx1 = bbb[n * 4 + 0];
            const float by1 = bbb[n * 4 + 1];
            const float bx2 = bbb[n * 4 + 2];
            const float by2 = bbb[n * 4 + 3];
            float ix = fminf(bx2, px1 + 16.0f) - fmaxf(bx1, px1);
            float iy = fminf(by2, py1 + 16.0f) - fmaxf(by1, py1);
            ix = fmaxf(ix, 0.0f);
            iy = fmaxf(iy, 0.0f);
            sR[idx] = ix * iy * (1.0f / 256.0f);
        }
    }
    __syncthreads();

    // ---- stage 2: R(32x196) x PT(196x768) via WMMA f32 16x16x4 ----
    const int wave = tid >> 5;
    const int lane = tid & 31;
    const int half = lane >> 4;   // 0: lanes 0-15, 1: lanes 16-31
    const int ln   = lane & 15;

    // A layout (16x4 f32): lanes 0-15 -> M=ln, VGPR{0,1}=K{0,1}; lanes 16-31 -> K{2,3}
    // B layout (4x16 f32): lanes 0-15 -> K={0,1} rows, lanes 16-31 -> K={2,3}, N=ln
    const float* ptb = pt + (size_t)b * NPATCH * HID;

    #pragma unroll
    for (int i = 0; i < 3; ++i) {
        const int  n0   = ((wave * 3 + i) + s * 24) * 16;   // N-tile base column
        const float* bp = ptb + n0 + ln;                    // per-lane column pointer
        v8f acc0 = {};   // M-tile 0 (rows 0-15)
        v8f acc1 = {};   // M-tile 1 (rows 16-31)

        for (int k = 0; k < NPATCH / 4; ++k) {
            const int k0 = 4 * k + 2 * half;    // this lane's first K row

            v2f Bv;
            Bv.x = bp[(size_t)(k0 + 0) * HID];
            Bv.y = bp[(size_t)(k0 + 1) * HID];

            if (k < 48)   // uniform guard; prefetch next k-step's rows (in-bounds)
                __builtin_prefetch(bp + (size_t)(4 * k + 4) * HID, 0, 0);

            // two contiguous f32 ratios -> ds_load_b64 (8B aligned: k0 even)
            const v2f A0 = *(const v2f*)&sR[(0  + ln) * NPATCH + k0];
            const v2f A1 = *(const v2f*)&sR[(16 + ln) * NPATCH + k0];

            acc0 = __builtin_amdgcn_wmma_f32_16x16x4_f32(
                       false, A0, false, Bv, (short)0, acc0, false, false);
            acc1 = __builtin_amdgcn_wmma_f32_16x16x4_f32(
                       false, A1, false, Bv, (short)0, acc1, false, false);
        }

        // ---- store D: VGPR v -> row (half*8 + v), col ln; rows offset by +1 (cls) ----
        const float inv = 1.0f / (float)NPATCH;
        float* ob = out + ((size_t)b * NTOK + 1) * HID + n0 + ln;
        #pragma unroll
        for (int v = 0; v < 8; ++v) {
            const int m = half * 8 + v;
            ob[(size_t)(m)      * HID] = acc0[v] * inv;
            ob[(size_t)(16 + m) * HID] = acc1[v] * inv;
        }
    }
}

extern "C" void kernel_launch(void* const* d_in, const int* in_sizes, int n_in,
                              void* d_out, int out_size, void* d_ws, size_t ws_size,
                              hipStream_t stream) {
    const float* cls  = (const float*)d_in[0];   // (B, 768)
    const float* pt   = (const float*)d_in[1];   // (B, 196, 768)
    const float* bbox = (const float*)d_in[2];   // (B, 32, 4)
    float*       out  = (float*)d_out;           // (B, 33, 768)

    const int B = in_sizes[0] / HID;
    roi_vit_wmma_kernel<<<dim3(B * NSPLIT), dim3(BLK), 0, stream>>>(cls, pt, bbox, out);
}